// PanopticHead_16415365005582
// MI455X (gfx1250) — compile-verified
//
#include <hip/hip_runtime.h>

#define STUFF 53
#define THING 80
#define MS    28
#define NINST 100
#define HH    256
#define WW    512
#define HWSZ  (HH * WW)

// ---------------------------------------------------------------------------
// Kernel 1: copy the 53 "stuff" planes verbatim (pure bandwidth, B128 moves).
// STUFF*HWSZ = 6,946,816 floats = 1,736,704 float4 = 6784 blocks * 256 threads.
// ---------------------------------------------------------------------------
__global__ __launch_bounds__(256)
void stuff_copy_kernel(const float* __restrict__ sem, float* __restrict__ out) {
    const int idx = blockIdx.x * 256 + threadIdx.x;
    const int n4 = (STUFF * HWSZ) / 4;
    if (idx < n4) {
        reinterpret_cast<float4*>(out)[idx] =
            reinterpret_cast<const float4*>(sem)[idx];
    }
}

// ---------------------------------------------------------------------------
// Kernel 2: per-instance bilinear mask paste + semantic add.
// grid = (NINST, HH/8); block = 256 threads (8 wave32 waves).
// Each block stages the instance's 28x28 mask (3136 B) into LDS with CDNA5
// async global->LDS DMA, then each thread produces 2 consecutive pixels for
// each of 8 rows (coalesced float2 stores).
// ---------------------------------------------------------------------------
__global__ __launch_bounds__(256)
void thing_paste_kernel(const float* __restrict__ mask_logits,
                        const float* __restrict__ sem,
                        const float* __restrict__ boxes,
                        const int*   __restrict__ classes,
                        float* __restrict__ out) {
    const int n    = blockIdx.x;      // instance
    const int rowg = blockIdx.y;      // 8-row group
    const int tid  = threadIdx.x;

    __shared__ float sm[MS * MS];     // 784 floats = 3136 bytes

    // ---- per-instance scalars (uniform across block) ----
    const int   cls = classes[n];
    const float bf0 = boxes[n * 4 + 0] * 0.25f;   // boxes / STRIDE (exact, pow2)
    const float bf1 = boxes[n * 4 + 1] * 0.25f;
    const float bf2 = boxes[n * 4 + 2] * 0.25f;
    const float bf3 = boxes[n * 4 + 3] * 0.25f;

    const int bx0 = (int)floorf(bf0);
    const int by0 = (int)floorf(bf1);
    const int bx1 = (int)floorf(bf2);
    const int by1 = (int)floorf(bf3);
    const int bw_i = bx1 - bx0 + 1;
    const int bh_i = by1 - by0 + 1;
    const float sxscale = (float)MS / (float)bw_i;
    const float syscale = (float)MS / (float)bh_i;

    // rounded "coarse" box (matches astype(int32) trunc and jnp.round = RNE)
    const int cx1 = (int)bf0;
    const int cy1 = (int)bf1;
    const int cx2 = (int)rintf(bf2) + 1;
    const int cy2 = (int)rintf(bf3) + 1;

    // ---- async stage mask tile into LDS (gfx1250 async-to-LDS DMA) ----
    {
        const float4* gsrc = reinterpret_cast<const float4*>(
            mask_logits + ((size_t)n * THING + (size_t)cls) * (MS * MS));
        // low 32 bits of a generic shared pointer == LDS byte offset (ISA 10.2)
        unsigned lds_base = (unsigned)(size_t)(void*)sm;
        if (tid < (MS * MS) / 4) {                 // 196 lanes issue one B128 each
            unsigned      laddr = lds_base + (unsigned)tid * 16u;
            const float4* gaddr = gsrc + tid;
            asm volatile("global_load_async_to_lds_b128 %0, %1, off"
                         :: "v"(laddr), "v"(gaddr) : "memory");
        }
        asm volatile("s_wait_asynccnt 0" ::: "memory");
        __syncthreads();
    }

    const float* semplane = sem + (size_t)(STUFF + cls) * HWSZ;
    float*       outplane = out + (size_t)(STUFF + n)   * HWSZ;

    // ---- x-direction coefficients: constant across the 8 rows ----
    const int  x   = tid * 2;                     // 256 threads * 2 px = 512
    const int  lxa = x - bx0;
    const int  lxb = x + 1 - bx0;
    const bool inxa = (lxa >= 0) && (lxa < bw_i);
    const bool inxb = (lxb >= 0) && (lxb < bw_i);

    float sxa = ((float)lxa + 0.5f) * sxscale - 0.5f;
    sxa = fminf(fmaxf(sxa, 0.0f), (float)(MS - 1));
    const int   xi0a = (int)sxa;
    const float wxa  = sxa - (float)xi0a;
    const int   xi1a = min(xi0a + 1, MS - 1);

    float sxb = ((float)lxb + 0.5f) * sxscale - 0.5f;
    sxb = fminf(fmaxf(sxb, 0.0f), (float)(MS - 1));
    const int   xi0b = (int)sxb;
    const float wxb  = sxb - (float)xi0b;
    const int   xi1b = min(xi0b + 1, MS - 1);

    const bool cinxa = (x     >= cx1) && (x     < cx2);
    const bool cinxb = (x + 1 >= cx1) && (x + 1 < cx2);

    // ---- 8 rows per block ----
    #pragma unroll
    for (int ry = 0; ry < 8; ++ry) {
        const int  y   = rowg * 8 + ry;
        const int  ly  = y - by0;
        const bool iny = (ly >= 0) && (ly < bh_i);

        float sy = ((float)ly + 0.5f) * syscale - 0.5f;
        sy = fminf(fmaxf(sy, 0.0f), (float)(MS - 1));
        const int   yi0 = (int)sy;
        const float wy  = sy - (float)yi0;
        const int   yi1 = min(yi0 + 1, MS - 1);

        const float* r0 = sm + yi0 * MS;
        const float* r1 = sm + yi1 * MS;

        float va = 0.0f, vb = 0.0f;
        if (iny && inxa) {
            const float m00 = r0[xi0a], m01 = r0[xi1a];
            const float m10 = r1[xi0a], m11 = r1[xi1a];
            const float top = (1.0f - wxa) * m00 + wxa * m01;
            const float bot = (1.0f - wxa) * m10 + wxa * m11;
            va = (1.0f - wy) * top + wy * bot;
        }
        if (iny && inxb) {
            const float m00 = r0[xi0b], m01 = r0[xi1b];
            const float m10 = r1[xi0b], m11 = r1[xi1b];
            const float top = (1.0f - wxb) * m00 + wxb * m01;
            const float bot = (1.0f - wxb) * m10 + wxb * m11;
            vb = (1.0f - wy) * top + wy * bot;
        }

        const bool ciny = (y >= cy1) && (y < cy2);
        const float sa = (ciny && cinxa) ? semplane[y * WW + x]     : 0.0f;
        const float sb = (ciny && cinxb) ? semplane[y * WW + x + 1] : 0.0f;

        float2 o;
        o.x = va + sa;
        o.y = vb + sb;
        *reinterpret_cast<float2*>(outplane + y * WW + x) = o;
    }
}

// ---------------------------------------------------------------------------
// Host launcher. Inputs (setup_inputs order):
//   d_in[0] mask_logits  f32 (100, 80, 28, 28)
//   d_in[1] sem_seg      f32 (133, 256, 512)
//   d_in[2] boxes        f32 (100, 4)
//   d_in[3] classes      i32 (100,)
// Output: f32 (1, 153, 256, 512)
// ---------------------------------------------------------------------------
extern "C" void kernel_launch(void* const* d_in, const int* in_sizes, int n_in,
                              void* d_out, int out_size, void* d_ws, size_t ws_size,
                              hipStream_t stream) {
    (void)in_sizes; (void)n_in; (void)out_size; (void)d_ws; (void)ws_size;

    const float* mask_logits = (const float*)d_in[0];
    const float* sem         = (const float*)d_in[1];
    const float* boxes       = (const float*)d_in[2];
    const int*   classes     = (const int*)d_in[3];
    float*       out         = (float*)d_out;

    const int n4 = (STUFF * HWSZ) / 4;                 // 1,736,704
    stuff_copy_kernel<<<dim3((n4 + 255) / 256), dim3(256), 0, stream>>>(sem, out);

    thing_paste_kernel<<<dim3(NINST, HH / 8), dim3(256), 0, stream>>>(
        mask_logits, sem, boxes, classes, out);
}